// SparsePool_43774306681224
// MI455X (gfx1250) — compile-verified
//
#include <hip/hip_runtime.h>

// SparsePool (segment-mean + gather) for MI455X / gfx1250.
// Memory-bound op: ~1.08 GB min traffic -> ~46us floor at 23.3 TB/s.
// Phase 1: zero counts+pooled workspace (L2-resident, 51.6 MB).
// Phase 2: wave-per-row scatter with global_atomic_add_f32 into L2.
// Phase 3: wave-per-row gather using CDNA5 async global->LDS loads
//          (global_load_async_to_lds_b128 / s_wait_asynccnt) with per-wave
//          double buffering, scale by 1/(count+eps), NT b128 stores.

#define SEGS 100000
#define FEAT 128
#define EPS_F 1e-9f

typedef float v4f __attribute__((ext_vector_type(4)));
typedef __attribute__((address_space(3))) float lds_f;

// Issue one 16B async copy from global memory to LDS (per active lane).
// ISA: GLOBAL_LOAD_ASYNC_TO_LDS_B128 — VDST = LDS byte address VGPR,
// VADDR = 64-bit global address VGPR pair, no SADDR ("off"). Tracked by ASYNCcnt.
__device__ __forceinline__ void async_copy16(float* lds_dst, const float* gsrc) {
    unsigned loff = (unsigned)(unsigned long long)(lds_f*)lds_dst;  // flat->LDS offset
    asm volatile("global_load_async_to_lds_b128 %0, %1, off"
                 :: "v"(loff), "v"(gsrc)
                 : "memory");
}

#define WAIT_ASYNCCNT(n) asm volatile("s_wait_asynccnt %0" :: "n"(n) : "memory")

__global__ void zero_ws_kernel(v4f* __restrict__ ws, int n4) {
    int i = blockIdx.x * blockDim.x + threadIdx.x;
    const int stride = gridDim.x * blockDim.x;
    v4f z = (v4f)0.0f;
    for (; i < n4; i += stride) ws[i] = z;   // RT stores: leave pooled hot in L2
}

__global__ void scatter_kernel(const float* __restrict__ x,
                               const int* __restrict__ idxv,
                               float* __restrict__ pooled,
                               float* __restrict__ counts,
                               int nrows) {
    const int lane = threadIdx.x & 31;
    const int wid  = blockIdx.x * (blockDim.x >> 5) + (threadIdx.x >> 5);
    const int nw   = gridDim.x * (blockDim.x >> 5);
    for (int row = wid; row < nrows; row += nw) {
        const int seg = idxv[row];
        // x is read exactly once: non-temporal b128 load keeps L2 for pooled.
        v4f v = __builtin_nontemporal_load((const v4f*)(x + (size_t)row * FEAT) + lane);
        float* dst = pooled + (size_t)seg * FEAT + lane * 4;
        atomicAdd(dst + 0, v.x);
        atomicAdd(dst + 1, v.y);
        atomicAdd(dst + 2, v.z);
        atomicAdd(dst + 3, v.w);
        if (lane == 0) atomicAdd(counts + seg, 1.0f);
    }
}

__global__ void gather_kernel(const float* __restrict__ pooled,
                              const float* __restrict__ counts,
                              const int* __restrict__ idxv,
                              float* __restrict__ out,
                              int nrows) {
    // 8 waves per 256-thread block; each wave owns a 2x512B double buffer.
    __shared__ float buf[8][2][FEAT];
    const int lane = threadIdx.x & 31;
    const int wsl  = threadIdx.x >> 5;
    const int wid  = blockIdx.x * 8 + wsl;
    const int nw   = gridDim.x * 8;

    int row = wid;
    int parity = 0;
    if (row < nrows) {
        int seg = idxv[row];
        async_copy16(&buf[wsl][0][lane * 4], pooled + (size_t)seg * FEAT + lane * 4);
    }
    while (row < nrows) {                 // wave-uniform control flow: EXEC all-1s
        const int next = row + nw;
        if (next < nrows) {
            int nseg = idxv[next];
            async_copy16(&buf[wsl][parity ^ 1][lane * 4],
                         pooled + (size_t)nseg * FEAT + lane * 4);
            WAIT_ASYNCCNT(1);             // async loads complete in order: row's data ready
        } else {
            WAIT_ASYNCCNT(0);
        }
        const int seg = idxv[row];
        const float inv = 1.0f / (counts[seg] + EPS_F);
        v4f v = *((const v4f*)&buf[wsl][parity][lane * 4]);
        v4f o = v * inv;
        // out is write-once 512MB: non-temporal b128 store, don't pollute L2.
        __builtin_nontemporal_store(o, (v4f*)(out + (size_t)row * FEAT) + lane);
        parity ^= 1;
        row = next;
    }
}

extern "C" void kernel_launch(void* const* d_in, const int* in_sizes, int n_in,
                              void* d_out, int out_size, void* d_ws, size_t ws_size,
                              hipStream_t stream) {
    (void)n_in; (void)out_size; (void)ws_size;
    const float* x     = (const float*)d_in[0];
    const int*   index = (const int*)d_in[1];
    float*       out   = (float*)d_out;
    const int    nrows = in_sizes[1];           // 1,000,000

    float* counts = (float*)d_ws;               // SEGS floats
    float* pooled = counts + SEGS;              // SEGS*FEAT floats (16B aligned: 400000 % 16 == 0)

    const int zero_n4 = (SEGS + SEGS * FEAT) / 4;
    zero_ws_kernel<<<2048, 256, 0, stream>>>((v4f*)d_ws, zero_n4);
    scatter_kernel<<<4096, 256, 0, stream>>>(x, index, pooled, counts, nrows);
    gather_kernel<<<4096, 256, 0, stream>>>(pooled, counts, index, out, nrows);
}